// LowRank2d_7284264534506
// MI455X (gfx1250) — compile-verified
//
#include <hip/hip_runtime.h>
#include <hip/hip_bf16.h>

// ---------------------------------------------------------------------------
// LowRank2d on MI455X (gfx1250), fp32 throughout via V_WMMA_F32_16X16X4_F32.
//
// Restructured computation (exact algebra, last linear layer pulled through
// the einsums):
//   h3psi = MLP_hidden(psi, a)        (16384 x 256)   [WMMA GEMMs]
//   G[b]  = h3psi[b]^T @ v[b]         (256 x 32)      [WMMA GEMM, A^T strides]
//   sv[b,i] = Sum_n v[b,n,i]
//   t[b,oir] = Sum_k W4psi[k,oir] G[b,k,i] + b4psi[oir] sv[b,i]   [scalar]
//   h3phi = MLP_hidden(phi, a)        (16384 x 256)   [WMMA GEMMs]
//   U[b,k,o] = Sum_ir W4phi[k,o*128+ir] t[b,o*128+ir]             [scalar]
//   c[b,o]   = Sum_ir b4phi[o*128+ir]  t[b,o*128+ir]              [scalar]
//   out[b]   = (h3phi[b] @ U[b] + c[b]) / N           [WMMA GEMM, bias+scale]
//
// Layer 1 (K=3) is zero-padded to K=4 so every GEMM instantiation takes the
// guard-free path. All strides (A, B, C) are compile-time so every load and
// store in the hot path uses base-pointer + immediate-offset addressing.
// ---------------------------------------------------------------------------

typedef float v2f __attribute__((ext_vector_type(2)));
typedef float v8f __attribute__((ext_vector_type(8)));

#define LR_BATCH 4
#define LR_N     4096
#define LR_MTOT  (LR_BATCH * LR_N)   // 16384 points
#define LR_W     32
#define LR_OIR   4096                // 32*32*4
#define LR_H3    256

// ---------------------------------------------------------------------------
// fp32 WMMA GEMM, all strides & K compile-time (KV % 4 == 0). One wave per
// (16*MT) x 16 output block: MT accumulator tiles share one B fragment per
// K-step (4 wmma : 4 A-loads : 2 B-loads when MT=4).
//   C[m,n] = act( Sum_k A[m*SAM + k*SAK] * B[k*LDB + n] + bias[n] ) * scale
// Fragment layouts per CDNA5 ISA 7.12.2 (f32 16x16x4):
//   A: lanes0-15 m=lane, v.x=K+0 v.y=K+1 ; lanes16-31 K+2, K+3
//   B: lanes0-15 n=lane, v.x=row K+0, v.y=row K+1 ; lanes16-31 rows K+2,K+3
//   C/D: vgpr j -> row m0 + j + 8*(lane>=16), col n0 + (lane&15)
// ---------------------------------------------------------------------------
#define LR_GEMM_KSTEP                                                         \
    {                                                                         \
        v2f bf = { pB[0], pB[LDB] };                                          \
        _Pragma("unroll")                                                     \
        for (int t = 0; t < MT; ++t) {                                        \
            v2f af;                                                           \
            if constexpr (SAK == 1 && (SAM % 2) == 0) {                       \
                af = *(const v2f*)pA[t];  /* contiguous pair -> b64 load */   \
            } else {                                                          \
                af = (v2f){ pA[t][0], pA[t][(long)SAK] };                     \
            }                                                                 \
            acc[t] = __builtin_amdgcn_wmma_f32_16x16x4_f32(                   \
                false, af, false, bf, (short)0, acc[t], false, false);        \
            pA[t] += 4 * (long)SAK;                                           \
        }                                                                     \
        pB += 4 * (long)LDB;                                                  \
    }

template<int SAM, int SAK, int LDB, int LDC, int KV, int MT,
         bool RELU, bool HASBIAS>
__global__ __launch_bounds__(32) void lr2d_wmma_gemm_f32(
    const float* __restrict__ A, long batchA,
    const float* __restrict__ B, long batchB,
    const float* __restrict__ bias, long batchBias,
    float* __restrict__ C, long batchC,
    float scale)
{
    static_assert((KV & 3) == 0, "KV must be a multiple of 4");

    const int lane = threadIdx.x;     // blockDim.x == 32 (wave32)
    const int half = lane >> 4;       // 0: lanes 0-15, 1: lanes 16-31
    const int l    = lane & 15;
    const int m0   = blockIdx.x * (16 * MT);
    const int n0   = blockIdx.y * 16;
    const long bz  = blockIdx.z;

    A += bz * batchA;
    B += bz * batchB;
    C += bz * batchC;

    v8f acc[MT];
    #pragma unroll
    for (int t = 0; t < MT; ++t) acc[t] = (v8f){0.f,0.f,0.f,0.f,0.f,0.f,0.f,0.f};

    const int ncol = n0 + l;
    const int kh   = 2 * half;                    // this lane-half's K offset

    const float* pB = B + (long)kh * LDB + ncol;
    const float* pA[MT];
    #pragma unroll
    for (int t = 0; t < MT; ++t)
        pA[t] = A + (long)(m0 + 16 * t + l) * SAM + (long)kh * SAK;

    if constexpr (KV <= 128) {
        #pragma unroll
        for (int k0 = 0; k0 < KV; k0 += 4) LR_GEMM_KSTEP
    } else {
        #pragma unroll 8
        for (int k0 = 0; k0 < KV; k0 += 4) LR_GEMM_KSTEP
    }

    // epilogue: vv = (acc + bias) * scale, optional ReLU before scale
    float bscaled = 0.f;
    if constexpr (HASBIAS) bscaled = (bias + bz * batchBias)[ncol];

    #pragma unroll
    for (int t = 0; t < MT; ++t) {
        float* pC = C + (long)(m0 + 16 * t + 8 * half) * LDC + ncol;
        #pragma unroll
        for (int j = 0; j < 8; ++j) {
            float vv = acc[t][j] + bscaled;
            if constexpr (RELU) vv = fmaxf(vv, 0.f);
            pC[j * LDC] = vv * scale;             // immediate-offset stores
        }
    }
}

// A4[m,0:4] = { a[m,0:3], 0 }  -- pad K=3 -> K=4 (shared by both MLPs)
__global__ __launch_bounds__(256) void lr2d_pad_a(
    const float* __restrict__ a, float* __restrict__ A4)
{
    const int idx = blockIdx.x * 256 + threadIdx.x;   // 0 .. 65535
    const int m = idx >> 2;
    const int j = idx & 3;
    A4[idx] = (j < 3) ? a[m * 3 + j] : 0.f;
}

// W04[k,0:64] = { W0[k,0:64] for k<3, 0 for k==3 }
__global__ __launch_bounds__(256) void lr2d_pad_w0(
    const float* __restrict__ W0, float* __restrict__ W4)
{
    const int idx = threadIdx.x;                      // 0 .. 255
    const int k = idx >> 6;
    W4[idx] = (k < 3) ? W0[idx] : 0.f;
}

// sv[b,i] = Sum_n v[b,n,i]   -- grid(4), block(32); coalesced 32-float rows
__global__ __launch_bounds__(32) void lr2d_colsum_v(
    const float* __restrict__ v, float* __restrict__ sv)
{
    const int b = blockIdx.x;
    const int i = threadIdx.x;
    const float* vb = v + (long)b * LR_N * LR_W;
    float acc = 0.f;
    for (int n = 0; n < LR_N; ++n) acc += vb[(long)n * LR_W + i];
    sv[b * LR_W + i] = acc;
}

// t[b,oir] = Sum_k W4psi[k,oir]*G[b,k,i] + b4psi[oir]*sv[b,i],  i=(oir>>2)&31
__global__ __launch_bounds__(256) void lr2d_compute_t(
    const float* __restrict__ W4psi, const float* __restrict__ b4psi,
    const float* __restrict__ G, const float* __restrict__ sv,
    float* __restrict__ t)
{
    const int idx = blockIdx.x * 256 + threadIdx.x;   // 0 .. 16383
    const int b   = idx >> 12;
    const int oir = idx & (LR_OIR - 1);
    const int i   = (oir >> 2) & (LR_W - 1);
    const float* Gb = G + (long)b * LR_H3 * LR_W + i;
    float acc = b4psi[oir] * sv[b * LR_W + i];
    for (int k = 0; k < LR_H3; ++k)
        acc += W4psi[(long)k * LR_OIR + oir] * Gb[(long)k * LR_W];
    t[idx] = acc;
}

// U[b,k,o] = Sum_ir W4phi[k, o*128+ir] * t[b, o*128+ir]
__global__ __launch_bounds__(256) void lr2d_compute_U(
    const float* __restrict__ W4phi, const float* __restrict__ t,
    float* __restrict__ U)
{
    const int idx = blockIdx.x * 256 + threadIdx.x;   // 0 .. 32767
    const int b   = idx >> 13;
    const int rem = idx & 8191;
    const int k   = rem >> 5;
    const int o   = rem & (LR_W - 1);
    const float* w  = W4phi + (long)k * LR_OIR + o * 128;
    const float* tb = t + (long)b * LR_OIR + o * 128;
    float acc = 0.f;
    #pragma unroll 4
    for (int ir = 0; ir < 128; ++ir) acc += w[ir] * tb[ir];
    U[((long)b * LR_H3 + k) * LR_W + o] = acc;
}

// c[b,o] = Sum_ir b4phi[o*128+ir] * t[b, o*128+ir]
__global__ __launch_bounds__(128) void lr2d_compute_c(
    const float* __restrict__ b4phi, const float* __restrict__ t,
    float* __restrict__ c)
{
    const int idx = threadIdx.x;          // 0 .. 127
    const int b = idx >> 5;
    const int o = idx & (LR_W - 1);
    const float* bb = b4phi + o * 128;
    const float* tb = t + (long)b * LR_OIR + o * 128;
    float acc = 0.f;
    #pragma unroll 4
    for (int ir = 0; ir < 128; ++ir) acc += bb[ir] * tb[ir];
    c[idx] = acc;
}

extern "C" void kernel_launch(void* const* d_in, const int* in_sizes, int n_in,
                              void* d_out, int out_size, void* d_ws, size_t ws_size,
                              hipStream_t stream)
{
    const float* v = (const float*)d_in[0];   // (4,4096,32)
    const float* a = (const float*)d_in[1];   // (4,4096,3)
    // phi params: W0 b0 W1 b1 W2 b2 W3 b3
    const float* pW0 = (const float*)d_in[2];
    const float* pb0 = (const float*)d_in[3];
    const float* pW1 = (const float*)d_in[4];
    const float* pb1 = (const float*)d_in[5];
    const float* pW2 = (const float*)d_in[6];
    const float* pb2 = (const float*)d_in[7];
    const float* pW3 = (const float*)d_in[8];
    const float* pb3 = (const float*)d_in[9];
    // psi params
    const float* qW0 = (const float*)d_in[10];
    const float* qb0 = (const float*)d_in[11];
    const float* qW1 = (const float*)d_in[12];
    const float* qb1 = (const float*)d_in[13];
    const float* qW2 = (const float*)d_in[14];
    const float* qb2 = (const float*)d_in[15];
    const float* qW3 = (const float*)d_in[16];
    const float* qb3 = (const float*)d_in[17];

    float* ws = (float*)d_ws;
    // workspace layout (floats)
    float* H3  = ws;                        // 16384*256 = 4,194,304 (psi then phi)
    float* H1  = ws + 4194304;              // 16384*64  = 1,048,576
    float* H2  = ws + 5242880;              // 16384*128 = 2,097,152
    float* G   = ws + 7340032;              // 4*256*32  = 32,768
    float* SV  = ws + 7372800;              // 4*32      = 128
    float* T   = ws + 7372928;              // 4*4096    = 16,384
    float* U   = ws + 7389312;              // 4*256*32  = 32,768
    float* CC  = ws + 7422080;              // 4*32      = 128
    float* A4  = ws + 7422208;              // 16384*4   = 65,536
    float* QW04= ws + 7487744;              // 4*64      = 256
    float* PW04= ws + 7488000;              // 4*64      = 256
    // total: 7,488,256 floats ~= 30 MB

    float* out = (float*)d_out;             // (4,4096,32)
    const dim3 wave(32);

    // ---- zero-pad layer-1 operands (K=3 -> K=4) ----
    lr2d_pad_a <<<dim3(LR_MTOT * 4 / 256), dim3(256), 0, stream>>>(a, A4);
    lr2d_pad_w0<<<dim3(1), dim3(256), 0, stream>>>(qW0, QW04);
    lr2d_pad_w0<<<dim3(1), dim3(256), 0, stream>>>(pW0, PW04);

    // ---- psi hidden MLP: A4 -> H1 -> H2 -> H3 (fused bias+ReLU GEMMs) ----
    // <SAM,SAK,LDB,LDC,KV,MT,RELU,HASBIAS>
    lr2d_wmma_gemm_f32<4, 1, 64, 64, 4, 4, true, true>
        <<<dim3(LR_MTOT/64, 4, 1), wave, 0, stream>>>(
        A4, 0,  QW04, 0,  qb0, 0,  H1, 0,  1.f);
    lr2d_wmma_gemm_f32<64, 1, 128, 128, 64, 4, true, true>
        <<<dim3(LR_MTOT/64, 8, 1), wave, 0, stream>>>(
        H1, 0,  qW1, 0,  qb1, 0,  H2, 0,  1.f);
    lr2d_wmma_gemm_f32<128, 1, 256, 256, 128, 4, true, true>
        <<<dim3(LR_MTOT/64, 16, 1), wave, 0, stream>>>(
        H2, 0,  qW2, 0,  qb2, 0,  H3, 0,  1.f);

    // ---- G[b] = h3psi[b]^T @ v[b]  (A accessed transposed via strides) ----
    lr2d_wmma_gemm_f32<1, LR_H3, LR_W, LR_W, LR_N, 4, false, false>
        <<<dim3(LR_H3/64, LR_W/16, LR_BATCH), wave, 0, stream>>>(
        H3, (long)LR_N * LR_H3,
        v,  (long)LR_N * LR_W,
        nullptr, 0,
        G, (long)LR_H3 * LR_W,
        1.f);

    // ---- sv, then t (pulls W4psi through the n-contraction) ----
    lr2d_colsum_v<<<dim3(LR_BATCH), dim3(32), 0, stream>>>(v, SV);
    lr2d_compute_t<<<dim3(LR_BATCH * LR_OIR / 256), dim3(256), 0, stream>>>(
        qW3, qb3, G, SV, T);

    // ---- phi hidden MLP (reuses H1/H2/H3 scratch) ----
    lr2d_wmma_gemm_f32<4, 1, 64, 64, 4, 4, true, true>
        <<<dim3(LR_MTOT/64, 4, 1), wave, 0, stream>>>(
        A4, 0,  PW04, 0,  pb0, 0,  H1, 0,  1.f);
    lr2d_wmma_gemm_f32<64, 1, 128, 128, 64, 4, true, true>
        <<<dim3(LR_MTOT/64, 8, 1), wave, 0, stream>>>(
        H1, 0,  pW1, 0,  pb1, 0,  H2, 0,  1.f);
    lr2d_wmma_gemm_f32<128, 1, 256, 256, 128, 4, true, true>
        <<<dim3(LR_MTOT/64, 16, 1), wave, 0, stream>>>(
        H2, 0,  pW2, 0,  pb2, 0,  H3, 0,  1.f);

    // ---- U, c (pull W4phi/b4phi through the ir-contraction) ----
    lr2d_compute_U<<<dim3(LR_BATCH * LR_H3 * LR_W / 256), dim3(256), 0, stream>>>(
        pW3, T, U);
    lr2d_compute_c<<<dim3(1), dim3(128), 0, stream>>>(pb3, T, CC);

    // ---- out[b] = (h3phi[b] @ U[b] + c[b]) / N ----
    lr2d_wmma_gemm_f32<LR_H3, 1, LR_W, LR_W, LR_H3, 4, false, true>
        <<<dim3(LR_N/64, LR_W/16, LR_BATCH), wave, 0, stream>>>(
        H3, (long)LR_N * LR_H3,
        U,  (long)LR_H3 * LR_W,
        CC, LR_W,
        out, (long)LR_N * LR_W,
        1.0f / (float)LR_N);
}